// StableHolographicLayer_88759794139153
// MI455X (gfx1250) — compile-verified
//
#include <hip/hip_runtime.h>
#include <hip/hip_bf16.h>
#include <math.h>

typedef __attribute__((ext_vector_type(2))) float v2f;
typedef __attribute__((ext_vector_type(8))) float v8f;

#define DIM   128
#define DIM2  256
#define EPSF  1e-5f

// ---- WMMA f32 16x16x4:  D(16x16) = A(16x4) * B(4x16) + C ----
__device__ __forceinline__ v8f wmma_f32_16x16x4(v2f a, v2f b, v8f c) {
  return __builtin_amdgcn_wmma_f32_16x16x4_f32(false, a, false, b, (short)0, c,
                                               false, false);
}

__device__ __forceinline__ float gelu_exact(float z) {
  return 0.5f * z * (1.0f + erff(z * 0.70710678118654752f));
}

// =====================  field value setup  =====================
__global__ void fv_kernel(const float* __restrict__ fb,   // [8][8][128]
                          const float* __restrict__ sm,   // [8][3][128]
                          float* __restrict__ fv)         // [8][128]
{
  int idx = blockIdx.x * blockDim.x + threadIdx.x;
  if (idx >= 8 * DIM) return;
  int i = idx >> 7;
  int c = idx & (DIM - 1);
  float px = (float)((i >> 2) & 1);
  float py = (float)((i >> 1) & 1);
  float pz = (float)(i & 1);
  float t = px * sm[(i * 3 + 0) * DIM + c]
          + py * sm[(i * 3 + 1) * DIM + c]
          + pz * sm[(i * 3 + 2) * DIM + c];
  float pw = 1.0f / (1.0f + expf(-t));
  float m = 0.f;
#pragma unroll
  for (int mm = 0; mm < 8; ++mm) m += fb[(i * 8 + mm) * DIM + c];
  fv[idx] = pw * m * 0.125f;
}

// =====================  per-module kernel  =====================
// Weights pair-interleaved in LDS: element (k,n) -> (k>>1)*STRIDE + 2n + (k&1),
// so each WMMA B operand {B[k][n], B[k+1][n]} is ONE conflict-free ds_load_b64
// (STRIDE % 64 == 32 banks keeps lane halves on disjoint bank halves).
//
// LDS layout (floats):
//   w1s  : 0      (64 kpairs * 544 = 34816)
//   w2s  : 34816  (64 kpairs * 288 = 18432)   <- one K-half of w2, re-staged mid-loop
//   b1s  : 53248 (256)  b2s : 53504 (128)
//   g1s  : 53632 (128)  be1s: 53760 (128)
//   g2s  : 53888 (128)  be2s: 54016 (128)  fvs: 54144 (128)
//   wave : 54272 + wave*2400 -> hb[16][132] (2112), gt[16][18] (288)
#define MOD_SMEM_FLOATS (54272 + 8 * 2400)

__global__ __launch_bounds__(256, 1)
void module_kernel(const float* __restrict__ x,
                   const float* __restrict__ fv,    // [128] (module slice)
                   float* __restrict__ acc,         // [B][128] running state sum
                   const float* __restrict__ w1,    // [128][256]
                   const float* __restrict__ b1,    // [256]
                   const float* __restrict__ w2,    // [256][128]
                   const float* __restrict__ b2,    // [128]
                   const float* __restrict__ g1,
                   const float* __restrict__ be1,
                   const float* __restrict__ g2,
                   const float* __restrict__ be2,
                   float nscale, int first)
{
  extern __shared__ float smem[];
  float* w1s  = smem;
  float* w2s  = smem + 34816;
  float* b1s  = smem + 53248;
  float* b2s  = smem + 53504;
  float* g1s  = smem + 53632;
  float* be1s = smem + 53760;
  float* g2s  = smem + 53888;
  float* be2s = smem + 54016;
  float* fvs  = smem + 54144;

  const int tid  = threadIdx.x;
  const int wave = tid >> 5;
  const int lane = tid & 31;
  const int nsel = lane & 15;
  const int half = lane >> 4;

  float* hb = smem + 54272 + wave * 2400;  // [16][132]
  float* gt = hb + 2112;                   // [16][18]

  // ---- stage w1 (pair-interleaved, kpair stride 544) ----
  {
    const float4* src = (const float4*)w1;
#pragma unroll
    for (int it = 0; it < 32; ++it) {
      int idx4 = tid + it * 256;            // 0..8191
      int k  = idx4 >> 6;                   // 64 float4 per 256-wide row
      int n4 = (idx4 & 63) << 2;
      float4 v = src[idx4];
      float* dst = w1s + (k >> 1) * 544 + (k & 1) + 2 * n4;
      dst[0] = v.x; dst[2] = v.y; dst[4] = v.z; dst[6] = v.w;
    }
  }
  // ---- stage w2 K-half 0 (pair-interleaved, kpair stride 288) ----
  {
    const float4* src = (const float4*)w2;
#pragma unroll
    for (int it = 0; it < 16; ++it) {
      int idx4 = tid + it * 256;            // 0..4095
      int kl = idx4 >> 5;                   // 32 float4 per 128-wide row
      int n4 = (idx4 & 31) << 2;
      float4 v = src[idx4];
      float* dst = w2s + (kl >> 1) * 288 + (kl & 1) + 2 * n4;
      dst[0] = v.x; dst[2] = v.y; dst[4] = v.z; dst[6] = v.w;
    }
  }
  b1s[tid] = b1[tid];
  if (tid < 128) {
    b2s[tid]  = b2[tid];
    g1s[tid]  = g1[tid];
    be1s[tid] = be1[tid];
    g2s[tid]  = g2[tid];
    be2s[tid] = be2[tid];
    fvs[tid]  = fv[tid];
  }
  __syncthreads();

  const int rowBase = blockIdx.x * 128 + wave * 16;

  // ---- phase 1: h = x + fv (+ nscale*acc); per-row stats latched in regs ----
  float mymean = 0.f, myvar = 0.f;
  {
    float4 f4 = *(const float4*)(fvs + 4 * lane);
    for (int r = 0; r < 16; ++r) {
      int row = rowBase + r;
      float4 xv = ((const float4*)x)[row * 32 + lane];
      float hx = xv.x + f4.x, hy = xv.y + f4.y,
            hz = xv.z + f4.z, hw = xv.w + f4.w;
      if (!first) {
        float4 av = ((const float4*)acc)[row * 32 + lane];
        hx += nscale * av.x; hy += nscale * av.y;
        hz += nscale * av.z; hw += nscale * av.w;
      }
      float4 h4; h4.x = hx; h4.y = hy; h4.z = hz; h4.w = hw;
      *(float4*)(hb + r * 132 + 4 * lane) = h4;
      float s = hx + hy + hz + hw;
      float q = hx * hx + hy * hy + hz * hz + hw * hw;
#pragma unroll
      for (int m = 16; m >= 1; m >>= 1) {
        s += __shfl_xor(s, m, 32);
        q += __shfl_xor(q, m, 32);
      }
      float mean = s * (1.0f / 128.0f);
      float var  = q * (1.0f / 128.0f) - mean * mean;
      if (r == nsel) { mymean = mean; myvar = var; }  // v_cndmask, no divergence
    }
  }
  float myrstd = rsqrtf(myvar + EPSF);

  // ---- A operand: fused LN1 while loading into registers (16x128 strip) ----
  v2f Areg[32];
  {
    int koff = 2 * half;
#pragma unroll
    for (int kk = 0; kk < 32; ++kk) {
      v2f raw = *(const v2f*)(hb + nsel * 132 + 4 * kk + koff);
      v2f gg  = *(const v2f*)(g1s + 4 * kk + koff);
      v2f bb  = *(const v2f*)(be1s + 4 * kk + koff);
      v2f a;
      a[0] = (raw[0] - mymean) * myrstd * gg[0] + bb[0];
      a[1] = (raw[1] - mymean) * myrstd * gg[1] + bb[1];
      Areg[kk] = a;
    }
  }

  // persistent GEMM2 accumulators (D layout), pre-biased with b2
  v8f acc2[8];
#pragma unroll
  for (int t2 = 0; t2 < 8; ++t2) {
    float bz = b2s[t2 * 16 + nsel];
#pragma unroll
    for (int v = 0; v < 8; ++v) acc2[t2][v] = bz;
  }

  // ---- main fused loop: two phases, re-stage w2 K-half between ----
  for (int ph = 0; ph < 2; ++ph) {
    if (ph == 1) {
      __syncthreads();   // all waves done with w2 half 0
      const float4* src = (const float4*)w2;
#pragma unroll
      for (int it = 0; it < 16; ++it) {
        int idx4 = tid + it * 256;
        int kl = idx4 >> 5;
        int n4 = (idx4 & 31) << 2;
        float4 v = src[(128 * 32) + idx4];   // rows 128..255
        float* dst = w2s + (kl >> 1) * 288 + (kl & 1) + 2 * n4;
        dst[0] = v.x; dst[2] = v.y; dst[4] = v.z; dst[6] = v.w;
      }
      __syncthreads();
    }
    for (int tt = 0; tt < 8; ++tt) {
      int t = ph * 8 + tt;
      v8f c1;
#pragma unroll
      for (int v = 0; v < 8; ++v) c1[v] = 0.f;
      int bcol2 = 2 * (t * 16 + nsel);
#pragma unroll
      for (int kk = 0; kk < 32; ++kk) {
        v2f b = *(const v2f*)(w1s + (2 * kk + half) * 544 + bcol2);
        c1 = wmma_f32_16x16x4(Areg[kk], b, c1);
      }
      // bias + exact GELU, stage 16x16 tile
      float bias = b1s[t * 16 + nsel];
#pragma unroll
      for (int v = 0; v < 8; ++v) {
        float z = c1[v] + bias;
        gt[(v + 8 * half) * 18 + nsel] = gelu_exact(z);
      }
      // consume tile as K-chunk of GEMM2
      v2f A2[4];
#pragma unroll
      for (int kk2 = 0; kk2 < 4; ++kk2)
        A2[kk2] = *(const v2f*)(gt + nsel * 18 + 4 * kk2 + 2 * half);
#pragma unroll
      for (int t2 = 0; t2 < 8; ++t2) {
        int n22 = 2 * (t2 * 16 + nsel);
#pragma unroll
        for (int kk2 = 0; kk2 < 4; ++kk2) {
          int lkp = tt * 8 + 2 * kk2 + half;   // local kpair in staged half
          v2f bb = *(const v2f*)(w2s + lkp * 288 + n22);
          acc2[t2] = wmma_f32_16x16x4(A2[kk2], bb, acc2[t2]);
        }
      }
    }
  }

  // ---- LN2 on D-layout registers ----
  {
    float sv[8], qv[8];
#pragma unroll
    for (int v = 0; v < 8; ++v) { sv[v] = 0.f; qv[v] = 0.f; }
#pragma unroll
    for (int t2 = 0; t2 < 8; ++t2)
#pragma unroll
      for (int v = 0; v < 8; ++v) {
        float z = acc2[t2][v];
        sv[v] += z; qv[v] += z * z;
      }
#pragma unroll
    for (int v = 0; v < 8; ++v) {
#pragma unroll
      for (int m = 8; m >= 1; m >>= 1) {
        sv[v] += __shfl_xor(sv[v], m, 32);
        qv[v] += __shfl_xor(qv[v], m, 32);
      }
    }
    // write normalized state into hb (D layout -> row-major), per-wave buffer
#pragma unroll
    for (int v = 0; v < 8; ++v) {
      float mean = sv[v] * (1.0f / 128.0f);
      float var  = qv[v] * (1.0f / 128.0f) - mean * mean;
      float rstd = rsqrtf(var + EPSF);
#pragma unroll
      for (int t2 = 0; t2 < 8; ++t2) {
        int c = t2 * 16 + nsel;
        hb[(v + 8 * half) * 132 + c] = (acc2[t2][v] - mean) * rstd * g2s[c] + be2s[c];
      }
    }
  }
  // coalesced float4 accumulate into acc (same-wave LDS ordering suffices)
  for (int r = 0; r < 16; ++r) {
    float4 v4 = *(const float4*)(hb + r * 132 + 4 * lane);
    int gi = (rowBase + r) * 32 + lane;
    if (first) {
      ((float4*)acc)[gi] = v4;
    } else {
      float4 old = ((const float4*)acc)[gi];
      v4.x += old.x; v4.y += old.y; v4.z += old.z; v4.w += old.w;
      ((float4*)acc)[gi] = v4;
    }
  }
}

// =====================  final combine kernel  =====================
// out = LN(acc/8; g,b) @ comb_w + comb_b  (in place on acc buffer)
// LDS: cws 64 kpairs * 288 = 18432 | cbs 128 | gs 128 | bes 128
//      per-wave 18816 + wave*2112 (hb only)
#define FIN_SMEM_FLOATS (18816 + 8 * 2112)

__global__ __launch_bounds__(256, 1)
void combine_kernel(float* __restrict__ accout,
                    const float* __restrict__ cw,   // [128][128]
                    const float* __restrict__ cb,
                    const float* __restrict__ g,
                    const float* __restrict__ be)
{
  extern __shared__ float smem[];
  float* cws = smem;            // pair-interleaved, kpair stride 288
  float* cbs = smem + 18432;
  float* gs  = smem + 18560;
  float* bes = smem + 18688;

  const int tid  = threadIdx.x;
  const int wave = tid >> 5;
  const int lane = tid & 31;
  const int nsel = lane & 15;
  const int half = lane >> 4;

  float* hb = smem + 18816 + wave * 2112;

  {
    const float4* src = (const float4*)cw;
#pragma unroll
    for (int it = 0; it < 16; ++it) {
      int idx4 = tid + it * 256;          // 0..4095
      int k  = idx4 >> 5;
      int n4 = (idx4 & 31) << 2;
      float4 v = src[idx4];
      float* dst = cws + (k >> 1) * 288 + (k & 1) + 2 * n4;
      dst[0] = v.x; dst[2] = v.y; dst[4] = v.z; dst[6] = v.w;
    }
  }
  if (tid < 128) { cbs[tid] = cb[tid]; gs[tid] = g[tid]; bes[tid] = be[tid]; }
  __syncthreads();

  const int rowBase = blockIdx.x * 128 + wave * 16;

  float mymean = 0.f, myvar = 0.f;
  for (int r = 0; r < 16; ++r) {
    float4 av = ((const float4*)accout)[(rowBase + r) * 32 + lane];
    float hx = av.x * 0.125f, hy = av.y * 0.125f,
          hz = av.z * 0.125f, hw = av.w * 0.125f;
    float4 h4; h4.x = hx; h4.y = hy; h4.z = hz; h4.w = hw;
    *(float4*)(hb + r * 132 + 4 * lane) = h4;
    float s = hx + hy + hz + hw;
    float q = hx * hx + hy * hy + hz * hz + hw * hw;
#pragma unroll
    for (int m = 16; m >= 1; m >>= 1) {
      s += __shfl_xor(s, m, 32);
      q += __shfl_xor(q, m, 32);
    }
    float mean = s * (1.0f / 128.0f);
    float var  = q * (1.0f / 128.0f) - mean * mean;
    if (r == nsel) { mymean = mean; myvar = var; }
  }
  float myrstd = rsqrtf(myvar + EPSF);

  // fused LN while loading A registers
  v2f Areg[32];
  {
    int koff = 2 * half;
#pragma unroll
    for (int kk = 0; kk < 32; ++kk) {
      v2f raw = *(const v2f*)(hb + nsel * 132 + 4 * kk + koff);
      v2f gg  = *(const v2f*)(gs + 4 * kk + koff);
      v2f bb  = *(const v2f*)(bes + 4 * kk + koff);
      v2f a;
      a[0] = (raw[0] - mymean) * myrstd * gg[0] + bb[0];
      a[1] = (raw[1] - mymean) * myrstd * gg[1] + bb[1];
      Areg[kk] = a;
    }
  }

  for (int t2 = 0; t2 < 8; ++t2) {
    float bz = cbs[t2 * 16 + nsel];
    v8f c;
#pragma unroll
    for (int v = 0; v < 8; ++v) c[v] = bz;
    int n22 = 2 * (t2 * 16 + nsel);
#pragma unroll
    for (int kk = 0; kk < 32; ++kk) {
      v2f b = *(const v2f*)(cws + (2 * kk + half) * 288 + n22);
      c = wmma_f32_16x16x4(Areg[kk], b, c);
    }
#pragma unroll
    for (int v = 0; v < 8; ++v)
      hb[(v + 8 * half) * 132 + t2 * 16 + nsel] = c[v];
  }
  // coalesced output store
  for (int r = 0; r < 16; ++r) {
    float4 v4 = *(const float4*)(hb + r * 132 + 4 * lane);
    ((float4*)accout)[(rowBase + r) * 32 + lane] = v4;
  }
}

// =====================  host launcher  =====================
extern "C" void kernel_launch(void* const* d_in, const int* in_sizes, int n_in,
                              void* d_out, int out_size, void* d_ws, size_t ws_size,
                              hipStream_t stream) {
  (void)n_in; (void)out_size; (void)ws_size;
  const float* x    = (const float*)d_in[0];
  const float* fb   = (const float*)d_in[1];
  const float* sm   = (const float*)d_in[2];
  const float* ln1g = (const float*)d_in[3];
  const float* ln1b = (const float*)d_in[4];
  const float* w1   = (const float*)d_in[5];
  const float* b1   = (const float*)d_in[6];
  const float* w2   = (const float*)d_in[7];
  const float* b2   = (const float*)d_in[8];
  const float* ln2g = (const float*)d_in[9];
  const float* ln2b = (const float*)d_in[10];
  const float* cg   = (const float*)d_in[11];
  const float* cbl  = (const float*)d_in[12];
  const float* cw   = (const float*)d_in[13];
  const float* cb   = (const float*)d_in[14];

  int B = in_sizes[0] / DIM;
  float* fv  = (float*)d_ws;       // 8*128 floats
  float* acc = (float*)d_out;      // accumulator doubles as output buffer

  fv_kernel<<<4, 256, 0, stream>>>(fb, sm, fv);

  size_t lds_mod = (size_t)MOD_SMEM_FLOATS * sizeof(float);
  hipFuncSetAttribute((const void*)module_kernel,
                      hipFuncAttributeMaxDynamicSharedMemorySize, (int)lds_mod);
  int blocks = B / 128;
  for (int i = 0; i < 8; ++i) {
    module_kernel<<<blocks, 256, lds_mod, stream>>>(
        x, fv + i * DIM, acc,
        w1 + (size_t)i * DIM * DIM2, b1 + (size_t)i * DIM2,
        w2 + (size_t)i * DIM2 * DIM, b2 + (size_t)i * DIM,
        ln1g + (size_t)i * DIM, ln1b + (size_t)i * DIM,
        ln2g + (size_t)i * DIM, ln2b + (size_t)i * DIM,
        i ? 0.1f / (float)i : 0.0f, i == 0 ? 1 : 0);
  }

  size_t lds_fin = (size_t)FIN_SMEM_FLOATS * sizeof(float);
  hipFuncSetAttribute((const void*)combine_kernel,
                      hipFuncAttributeMaxDynamicSharedMemorySize, (int)lds_fin);
  combine_kernel<<<blocks, 256, lds_fin, stream>>>(acc, cw, cb, cg, cbl);
}